// Conv2d_48533130445551
// MI455X (gfx1250) — compile-verified
//
#include <hip/hip_runtime.h>

typedef __attribute__((ext_vector_type(2))) float v2f;
typedef __attribute__((ext_vector_type(8))) float v8f;

#define BATCH   16
#define CIN     128
#define COUT    256
#define HW      56
#define SPATIAL (HW * HW)          // 3136
#define NPIX    (BATCH * SPATIAL)  // 50176
#define KTOT    (CIN * 9)          // 1152
#define KC      16                 // K chunk per LDS stage
#define KP      (KC / 2)           // float2 K-pairs per stage
#define MT      64                 // couts per workgroup
#define NT      128                // pixels per workgroup
#define A2S     80                 // float2 stride; 80 % 32 == 16 -> halves on disjoint banks
#define B2S     144                // float2 stride; 144 % 32 == 16

__global__ __launch_bounds__(256) void conv3x3_wmma_f32(
    const float* __restrict__ x, const float* __restrict__ w,
    const float* __restrict__ bias, float* __restrict__ out)
{
    // K-pair interleaved layout: element [kp][c] = {M[2kp][c], M[2kp+1][c]}
    // -> every WMMA fragment is one aligned ds_load_b64, no repacking movs.
    __shared__ v2f a2_s[2][KP * A2S];   // [buf][kp][cout_local]
    __shared__ v2f b2_s[2][KP * B2S];   // [buf][kp][pixel_local]

    const int tid    = threadIdx.x;
    const int lane   = tid & 31;
    const int wave   = tid >> 5;
    const int m_base = blockIdx.y * MT;
    const int n_base = blockIdx.x * NT;
    const int wm     = wave & 3;   // M subtile (16 couts)
    const int wn     = wave >> 2;  // N half (64 pixels)

    // A staging: thread -> (cout_local, k-quad), one float4 global load
    const int a_cl = tid >> 2;           // 0..63
    const int a_kq = (tid & 3) << 2;     // 0,4,8,12
    const float* wptr = w + (size_t)(m_base + a_cl) * KTOT + a_kq;

    // B staging: thread -> k-row (0..15), 8 consecutive pixels.
    const int b_r  = tid >> 4;           // 0..15
    const int b_pg = (tid & 15) << 3;    // 0..120
    int pbase[8], poh[8], pow_[8];
#pragma unroll
    for (int j = 0; j < 8; ++j) {
        int pn = n_base + b_pg + j;
        int bb = pn / SPATIAL;
        int s  = pn - bb * SPATIAL;
        int oh = s / HW;
        int ow = s - oh * HW;
        pbase[j] = bb * (CIN * SPATIAL) + oh * HW + ow;
        poh[j]   = oh;
        pow_[j]  = ow;
    }

    // im2col gather for k-rows [kkx+b_r] -> 8 register values
    auto gather = [&](int kkx, float* bv) {
        int k    = kkx + b_r;
        int ci   = k / 9;
        int rem  = k - ci * 9;
        int kh   = rem / 3;
        int kw   = rem - kh * 3;
        int coff = ci * SPATIAL + (kh - 1) * HW + (kw - 1);
#pragma unroll
        for (int j = 0; j < 8; ++j) {
            int ih = poh[j] + kh - 1;
            int iw = pow_[j] + kw - 1;
            bool ok = ((unsigned)ih < (unsigned)HW) &&
                      ((unsigned)iw < (unsigned)HW);
            bv[j] = ok ? x[pbase[j] + coff] : 0.f;
        }
    };

    auto commit = [&](int buf, const float4& w4, const float* bv) {
        v2f* a2 = a2_s[buf];
        v2f av0; av0.x = w4.x; av0.y = w4.y;
        v2f av1; av1.x = w4.z; av1.y = w4.w;
        a2[((a_kq >> 1) + 0) * A2S + a_cl] = av0;
        a2[((a_kq >> 1) + 1) * A2S + a_cl] = av1;
        float* bf = (float*)b2_s[buf];
        int bbase = ((b_r >> 1) * B2S) * 2 + (b_r & 1);
#pragma unroll
        for (int j = 0; j < 8; ++j)
            bf[bbase + (b_pg + j) * 2] = bv[j];
    };

    v8f acc[4];
#pragma unroll
    for (int i = 0; i < 4; ++i) acc[i] = (v8f)0.f;

    const int row    = lane & 15;
    const int kphalf = lane >> 4;              // lane-half selects K pair {0,1} vs {2,3}
    const int a_col  = wm * 16 + row;
    const int b_col0 = wn * 64 + row;

    // ---- prologue: stage chunk 0 ----
    {
        float4 w4 = *(const float4*)(wptr);
        float bv[8];
        gather(0, bv);
        commit(0, w4, bv);
    }
    __syncthreads();

    for (int kk = 0; kk < KTOT; kk += KC) {
        const int  buf  = (kk / KC) & 1;
        const bool more = (kk + KC) < KTOT;

        // issue next chunk's global loads before compute (pipeline)
        float4 nw4;
        float  nbv[8];
        if (more) {
            nw4 = *(const float4*)(wptr + kk + KC);
            gather(kk + KC, nbv);
        }

        // ---- compute: 4 K-steps x 4 N-tiles of V_WMMA_F32_16X16X4_F32 ----
        const v2f* a2 = a2_s[buf];
        const v2f* b2 = b2_s[buf];
#pragma unroll
        for (int ks = 0; ks < 4; ++ks) {
            int kp = 2 * ks + kphalf;
            v2f a = a2[kp * A2S + a_col];
#pragma unroll
            for (int nt = 0; nt < 4; ++nt) {
                v2f b = b2[kp * B2S + b_col0 + nt * 16];
                acc[nt] = __builtin_amdgcn_wmma_f32_16x16x4_f32(
                    false, a, false, b, (short)0, acc[nt], false, false);
            }
        }

        if (more) commit(buf ^ 1, nw4, nbv);
        __syncthreads();
    }

    // ---- epilogue: D layout -> cout = m0 + v + 8*(lane>=16), pixel = lane%16
    const int mrow0 = m_base + wm * 16 + kphalf * 8;
#pragma unroll
    for (int nt = 0; nt < 4; ++nt) {
        int pn = n_base + wn * 64 + nt * 16 + row;
        int bb = pn / SPATIAL;
        int s  = pn - bb * SPATIAL;
        size_t obase = (size_t)bb * (COUT * SPATIAL) + s;
#pragma unroll
        for (int v = 0; v < 8; ++v) {
            int cout = mrow0 + v;
            __builtin_nontemporal_store(acc[nt][v] + bias[cout],
                                        &out[obase + (size_t)cout * SPATIAL]);
        }
    }
}

extern "C" void kernel_launch(void* const* d_in, const int* in_sizes, int n_in,
                              void* d_out, int out_size, void* d_ws, size_t ws_size,
                              hipStream_t stream) {
    const float* x    = (const float*)d_in[0];
    const float* wgt  = (const float*)d_in[1];
    const float* bias = (const float*)d_in[2];
    float* out        = (float*)d_out;

    dim3 grid(NPIX / NT, COUT / MT, 1);   // 392 x 4
    dim3 block(256, 1, 1);                // 8 waves (wave32)
    conv3x3_wmma_f32<<<grid, block, 0, stream>>>(x, wgt, bias, out);
}